// VectorQuantizer_32538672234736
// MI455X (gfx1250) — compile-verified
//
#include <hip/hip_runtime.h>
#include <hip/hip_bf16.h>

typedef __attribute__((ext_vector_type(2))) float v2f;
typedef __attribute__((ext_vector_type(8))) float v8f;

#define EMB_DIM 64
#define NUM_CODES 1024
#define M_CHAINS 4                     // 16-row strips per wave

// ---------------------------------------------------------------------------
// Kernel 1: ebias[c] = 0.5 * ||emb[c]||^2
// ---------------------------------------------------------------------------
__global__ void vq_ebias_kernel(const float* __restrict__ emb,
                                float* __restrict__ ebias) {
    int c = blockIdx.x * blockDim.x + threadIdx.x;
    if (c < NUM_CODES) {
        const float* e = emb + c * EMB_DIM;
        float s = 0.0f;
#pragma unroll
        for (int d = 0; d < EMB_DIM; d += 4) {
            float4 v = *(const float4*)(e + d);
            s += v.x * v.x + v.y * v.y + v.z * v.z + v.w * v.w;
        }
        ebias[c] = 0.5f * s;
    }
}

// ---------------------------------------------------------------------------
// Kernel 2: per-row argmax of (z . e_n - 0.5*||e_n||^2) via f32 WMMA.
// One wave handles a 64-row strip: four 16-row A panels share every B
// fragment -> 64 WMMA per 16 B-loads (4:1), quartering WGP$ emb traffic and
// giving four independent accumulation chains for deep pipelining.
// Index for row r is written (as int bits) into outIdx[r*64 + 63], i.e. inside
// the row's own output segment (overwritten later by the gather kernel).
// ---------------------------------------------------------------------------
__global__ void __launch_bounds__(128)
vq_argmin_kernel(const float* __restrict__ z,
                 const float* __restrict__ emb,
                 const float* __restrict__ ebias,
                 int* __restrict__ outIdx) {
    const int lane = threadIdx.x & 31;
    const int lo   = lane & 15;        // row / column within 16x16 tile
    const int hi   = (lane >> 4) & 1;  // selects K-pair {0,1} vs {2,3}
    const int wave = blockIdx.x * (blockDim.x >> 5) + (threadIdx.x >> 5);
    const int rowBase = wave * (16 * M_CHAINS);

    // A panels: rows [rowBase + 16*c, rowBase + 16*(c+1)) for c = 0..3.
    // f32 16x4 A layout: lane%16 = M row; v0/v1 hold K = 2*hi, 2*hi+1.
    v2f a[M_CHAINS][16];
#pragma unroll
    for (int c = 0; c < M_CHAINS; ++c) {
        const float* zrow = z + (size_t)(rowBase + 16 * c + lo) * EMB_DIM + 2 * hi;
#pragma unroll
        for (int s = 0; s < 16; ++s)
            a[c][s] = *(const v2f*)(zrow + 4 * s);
    }

    float bestV[M_CHAINS][8];
    int   bestI[M_CHAINS][8];
#pragma unroll
    for (int c = 0; c < M_CHAINS; ++c)
#pragma unroll
        for (int v = 0; v < 8; ++v) { bestV[c][v] = -3.402823466e38f; bestI[c][v] = 0; }

    for (int t = 0; t < NUM_CODES / 16; ++t) {
        const int code = t * 16 + lo;          // column n this lane owns
        const float nbias = -ebias[code];

        v8f acc[M_CHAINS];
#pragma unroll
        for (int c = 0; c < M_CHAINS; ++c)
#pragma unroll
            for (int i = 0; i < 8; ++i) acc[c][i] = nbias;   // C = -0.5*||e_n||^2

        // B panel fragment pattern mirrors A (B[k][n] = emb[n][k]).
        const float* erow = emb + (size_t)code * EMB_DIM + 2 * hi;
#pragma unroll
        for (int s = 0; s < 16; ++s) {
            v2f b = *(const v2f*)(erow + 4 * s);
#pragma unroll
            for (int c = 0; c < M_CHAINS; ++c)
                acc[c] = __builtin_amdgcn_wmma_f32_16x16x4_f32(
                    false, a[c][s], false, b, (short)0, acc[c], false, false);
        }

        // acc[c][v] = score for row rowBase + 16*c + v + 8*hi, code n = code.
#pragma unroll
        for (int c = 0; c < M_CHAINS; ++c)
#pragma unroll
            for (int v = 0; v < 8; ++v) {
                float sc = acc[c][v];
                if (sc > bestV[c][v]) { bestV[c][v] = sc; bestI[c][v] = code; }
            }
    }

    // Cross-lane argmax within each 16-lane half (xor 1,2,4,8 stays in-half).
    // Tie -> lower index (matches jnp.argmin first-min semantics).
#pragma unroll
    for (int c = 0; c < M_CHAINS; ++c)
#pragma unroll
        for (int v = 0; v < 8; ++v) {
            float bv = bestV[c][v];
            int   bi = bestI[c][v];
#pragma unroll
            for (int m = 1; m < 16; m <<= 1) {
                float ov = __shfl_xor(bv, m, 32);
                int   oi = __shfl_xor(bi, m, 32);
                if (ov > bv || (ov == bv && oi < bi)) { bv = ov; bi = oi; }
            }
            bestV[c][v] = bv; bestI[c][v] = bi;
        }

    if (lo == 0) {
#pragma unroll
        for (int c = 0; c < M_CHAINS; ++c)
#pragma unroll
            for (int v = 0; v < 8; ++v) {
                int row = rowBase + 16 * c + v + 8 * hi;
                outIdx[(size_t)row * EMB_DIM + 63] = bestI[c][v];
            }
    }
}

// ---------------------------------------------------------------------------
// Kernel 3: gather quantized = emb[idx], accumulate sum((q - z)^2).
// Block = 256 threads = 4 rows x 64 dims. Index read before the row segment
// is overwritten; only the owning block touches its rows.
// ---------------------------------------------------------------------------
__global__ void __launch_bounds__(256)
vq_gather_loss_kernel(const float* __restrict__ z,
                      const float* __restrict__ emb,
                      float* __restrict__ out,
                      double* __restrict__ acc) {
    __shared__ float red[256];
    const int tx  = threadIdx.x & 63;
    const int ty  = threadIdx.x >> 6;
    const int row = blockIdx.x * 4 + ty;

    const int idx = ((const int*)out)[(size_t)row * EMB_DIM + 63];
    __syncthreads();   // all 4 rows' indices read before any overwrite

    float q  = emb[(size_t)idx * EMB_DIM + tx];
    float zz = z[(size_t)row * EMB_DIM + tx];
    out[(size_t)row * EMB_DIM + tx] = q;

    float d = q - zz;
    red[threadIdx.x] = d * d;
    __syncthreads();
#pragma unroll
    for (int s = 128; s > 0; s >>= 1) {
        if (threadIdx.x < s) red[threadIdx.x] += red[threadIdx.x + s];
        __syncthreads();
    }
    if (threadIdx.x == 0) atomicAdd(acc, (double)red[0]);
}

// ---------------------------------------------------------------------------
// Kernel 4: loss = (1 + 0.25) * sum / N_total
// ---------------------------------------------------------------------------
__global__ void vq_finalize_kernel(const double* __restrict__ acc,
                                   float* __restrict__ out, int ntot) {
    out[ntot] = (float)(1.25 * acc[0] / (double)ntot);
}

// ---------------------------------------------------------------------------
extern "C" void kernel_launch(void* const* d_in, const int* in_sizes, int n_in,
                              void* d_out, int out_size, void* d_ws, size_t ws_size,
                              hipStream_t stream) {
    const float* z   = (const float*)d_in[0];   // [N, 64]
    const float* emb = (const float*)d_in[1];   // [1024, 64]
    float* out = (float*)d_out;                 // [N*64] quantized + [1] loss

    const int ntot = in_sizes[0];               // 8388608
    const int rows = ntot / EMB_DIM;            // 131072

    double* acc  = (double*)d_ws;                       // 8B accumulator
    float* ebias = (float*)((char*)d_ws + 16);          // 1024 floats

    hipMemsetAsync(d_ws, 0, 16, stream);

    vq_ebias_kernel<<<(NUM_CODES + 255) / 256, 256, 0, stream>>>(emb, ebias);

    // 1 wave per 64 rows; 128-thread blocks = 4 waves = 256 rows per block.
    vq_argmin_kernel<<<rows / (64 * M_CHAINS / 4 * 4), 128, 0, stream>>>
        (z, emb, ebias, (int*)d_out);

    // 4 rows per 256-thread block.
    vq_gather_loss_kernel<<<rows / 4, 256, 0, stream>>>(z, emb, out, acc);

    vq_finalize_kernel<<<1, 1, 0, stream>>>(acc, out, ntot);
}